// EdgeConvStack_89489938579858
// MI455X (gfx1250) — compile-verified
//
#include <hip/hip_runtime.h>
#include <cstdint>

typedef __attribute__((ext_vector_type(2))) float v2f;
typedef __attribute__((ext_vector_type(8))) float v8f;

#define KNN_K     20
#define NEG_SLOPE 0.2f
#define BN_EPS    1e-5f
#define TOTC      256   // 64 + 64 + 128 output channels

// ---------------------------------------------------------------------------
// Layer-0 prep: transpose x (B,C,N) -> xT (B,N,CPAD) with zero-padded
// channels, and fuse the squared-norm xx[b][n]. After this, every WMMA
// operand load in knn/ptrans is an unguarded contiguous v2f (b64) load.
// ---------------------------------------------------------------------------
__global__ void prep_x_kernel(const float* __restrict__ x, float* __restrict__ xT,
                              float* __restrict__ xx, int C, int CPAD, int N, int B) {
  int gid = blockIdx.x * blockDim.x + threadIdx.x;
  if (gid >= B * N) return;
  int b = gid / N, n = gid - b * N;
  float s = 0.f;
  for (int c = 0; c < CPAD; ++c) {
    float v = (c < C) ? x[((size_t)b * C + c) * N + n] : 0.f;
    xT[(size_t)gid * CPAD + c] = v;
    s += v * v;
  }
  xx[gid] = s;
}

// ---------------------------------------------------------------------------
// Pad/split weights: W1p[o][c] = W[o][c], Wdp[o][c] = W[o][C+c] - W[o][c]
// (zero padded to CPAD). Then U = W1p @ x (gathered via idx) and
// V = Wdp @ x accumulate directly in WMMA with no epilogue subtract.
// ---------------------------------------------------------------------------
__global__ void prep_w_kernel(const float* __restrict__ W, float* __restrict__ W1p,
                              float* __restrict__ Wdp, int C, int CPAD, int O) {
  int gid = blockIdx.x * blockDim.x + threadIdx.x;
  if (gid >= O * CPAD) return;
  int o = gid / CPAD, c = gid - o * CPAD;
  float w1 = (c < C) ? W[(size_t)o * 2 * C + c] : 0.f;
  float w2 = (c < C) ? W[(size_t)o * 2 * C + C + c] : 0.f;
  W1p[gid] = w1;
  Wdp[gid] = w2 - w1;
}

// ---------------------------------------------------------------------------
// Fused kNN: 16x16 score tiles via V_WMMA_F32_16X16X4_F32, top-20 kept in
// registers by 16 worker lanes per wave. Score = inner(n,m) - 0.5*||x_m||^2
// (the row-constant -||x_n||^2 term does not change per-row ranking; the
// column term is folded into the WMMA accumulator init).
//
// WMMA f32 16x16x4 register layout (CDNA5 ISA 7.12.2):
//   A (16x4): lane l -> row M = l%16; VGPR0/1 = K = {0,1} (lanes 0-15),
//             {2,3} (lanes 16-31)
//   B (4x16): lane l -> col N = l%16; VGPR0/1 = K = {0,1}/{2,3} by lane half
//   C/D     : VGPR v, lane l -> (M = v + 8*(l/16), N = l%16)
// ---------------------------------------------------------------------------
template <int CPAD>
__global__ void __launch_bounds__(256)
knn_kernel(const float* __restrict__ xT, const float* __restrict__ xx,
           int* __restrict__ knn, int N, int B) {
  __shared__ float sdist[8][256];                 // one 16x16 tile per wave
  const int lane = threadIdx.x & 31;
  const int wave = threadIdx.x >> 5;
  const int tilesPerB = N >> 4;
  const int gtile = blockIdx.x * (blockDim.x >> 5) + wave;   // grid sized exactly
  const int b  = gtile / tilesPerB;
  const int q0 = (gtile - b * tilesPerB) << 4;
  const int half = lane >> 4;                     // 0: K={0,1}, 1: K={2,3}
  const int mrow = lane & 15;
  const float* xTb = xT + (size_t)b * N * CPAD;
  const float* xxb = xx + (size_t)b * N;

  // Preload A tile (16 query points x CPAD channels) as contiguous v2f loads.
  v2f a[CPAD / 4];
  {
    const float* aq = xTb + (size_t)(q0 + mrow) * CPAD + half * 2;
#pragma unroll
    for (int kc = 0; kc < CPAD / 4; ++kc)
      a[kc] = *(const v2f*)(aq + kc * 4);
  }

  float bd[KNN_K]; int bi[KNN_K];
#pragma unroll
  for (int i = 0; i < KNN_K; ++i) { bd[i] = -3.0e38f; bi[i] = 0; }

  const int ntiles = N >> 4;
  for (int t = 0; t < ntiles; ++t) {
    const int cand = (t << 4) + mrow;             // candidate column this lane owns
    const float cinit = -0.5f * xxb[cand];        // fold -0.5*||x_m||^2 into C
    v8f acc;
#pragma unroll
    for (int i = 0; i < 8; ++i) acc[i] = cinit;

    const float* xc = xTb + (size_t)cand * CPAD + half * 2;
#pragma unroll
    for (int kc = 0; kc < CPAD / 4; ++kc) {
      v2f bm = *(const v2f*)(xc + kc * 4);
      acc = __builtin_amdgcn_wmma_f32_16x16x4_f32(false, a[kc], false, bm,
                                                  (short)0, acc, false, false);
    }
    if (t + 1 < ntiles)
      __builtin_prefetch(xTb + (size_t)(cand + 16) * CPAD, 0, 3);

    // Spill tile to LDS in (row, col) order.
#pragma unroll
    for (int v = 0; v < 8; ++v)
      sdist[wave][((v + (half << 3)) << 4) + mrow] = acc[v];
    __syncthreads();

    if (lane < 16) {                              // lane owns query row q0+lane
#pragma unroll
      for (int j = 0; j < 16; ++j) {
        const float d = sdist[wave][(lane << 4) + j];
        if (d > bd[KNN_K - 1]) {                  // beats current minimum
          bd[KNN_K - 1] = d; bi[KNN_K - 1] = (t << 4) + j;
#pragma unroll
          for (int q = KNN_K - 1; q > 0; --q) {   // bubble into sorted list
            if (bd[q] > bd[q - 1]) {
              const float td = bd[q]; bd[q] = bd[q - 1]; bd[q - 1] = td;
              const int   ti = bi[q]; bi[q] = bi[q - 1]; bi[q - 1] = ti;
            }
          }
        }
      }
    }
    __syncthreads();
  }

  if (lane < 16) {
    int* op = knn + ((size_t)b * N + q0 + lane) * KNN_K;
#pragma unroll
    for (int i = 0; i < KNN_K; ++i) op[i] = bi[i];
  }
}

// ---------------------------------------------------------------------------
// Point transform via WMMA GEMM:
//   U[b,o,n] = W1p @ x,   V[b,o,n] = (W2-W1)p @ x
// so that the edge-conv output is y[b,o,n,k] = U[b,o,idx(n,k)] + V[b,o,n].
// ---------------------------------------------------------------------------
template <int CPAD>
__global__ void __launch_bounds__(256)
ptrans_kernel(const float* __restrict__ xT, const float* __restrict__ W1p,
              const float* __restrict__ Wdp, float* __restrict__ U,
              float* __restrict__ V, int O, int N, int B) {
  const int lane = threadIdx.x & 31;
  const int wave = threadIdx.x >> 5;
  const int gw = blockIdx.x * (blockDim.x >> 5) + wave;
  const int ntiles = N >> 4, otiles = O >> 4;
  const int b   = gw / (ntiles * otiles);
  const int rem = gw - b * (ntiles * otiles);
  const int o0  = (rem / ntiles) << 4;
  const int n0  = (rem - (rem / ntiles) * ntiles) << 4;
  const int half = lane >> 4;
  const int mrow = lane & 15;

  const float* w1 = W1p + (size_t)(o0 + mrow) * CPAD + half * 2;
  const float* wd = Wdp + (size_t)(o0 + mrow) * CPAD + half * 2;
  const float* xr = xT + ((size_t)b * N + n0 + mrow) * CPAD + half * 2;

  v8f cu = {};
  v8f cv = {};
#pragma unroll
  for (int kc = 0; kc < CPAD / 4; ++kc) {
    v2f a1 = *(const v2f*)(w1 + kc * 4);
    v2f a2 = *(const v2f*)(wd + kc * 4);
    v2f bm = *(const v2f*)(xr + kc * 4);
    cu = __builtin_amdgcn_wmma_f32_16x16x4_f32(false, a1, false, bm, (short)0, cu, false, false);
    cv = __builtin_amdgcn_wmma_f32_16x16x4_f32(false, a2, false, bm, (short)0, cv, false, false);
  }
#pragma unroll
  for (int v = 0; v < 8; ++v) {
    const int o = o0 + v + (half << 3);
    const size_t off = ((size_t)b * O + o) * N + n0 + mrow;
    U[off] = cu[v];
    V[off] = cv[v];
  }
}

// ---------------------------------------------------------------------------
// Per-channel sum / sum-of-squares of y = U[idx] + V (y never materialized).
// stats[o] = sum, stats[128+o] = sumsq
// ---------------------------------------------------------------------------
__global__ void __launch_bounds__(256)
bn_stats_kernel(const float* __restrict__ U, const float* __restrict__ V,
                const int* __restrict__ knn, float* __restrict__ stats,
                int O, int N, int B) {
  const int o = blockIdx.x;
  const int tid = threadIdx.x;
  float s1 = 0.f, s2 = 0.f;
  const int bnTotal = B * N;
  for (int bn = blockIdx.y * blockDim.x + tid; bn < bnTotal;
       bn += gridDim.y * blockDim.x) {
    const int b = bn / N, n = bn - b * N;
    const float* Ub = U + ((size_t)b * O + o) * N;
    const float vv = V[((size_t)b * O + o) * N + n];
    const int* ip = knn + (size_t)bn * KNN_K;
#pragma unroll
    for (int k = 0; k < KNN_K; ++k) {
      const float y = Ub[ip[k]] + vv;
      s1 += y; s2 += y * y;
    }
  }
  __shared__ float sh1[256];
  __shared__ float sh2[256];
  sh1[tid] = s1; sh2[tid] = s2;
  __syncthreads();
  for (int st = 128; st > 0; st >>= 1) {
    if (tid < st) { sh1[tid] += sh1[tid + st]; sh2[tid] += sh2[tid + st]; }
    __syncthreads();
  }
  if (tid == 0) {
    atomicAdd(&stats[o], sh1[0]);
    atomicAdd(&stats[128 + o], sh2[0]);
  }
}

__global__ void bn_finalize_kernel(const float* __restrict__ stats,
                                   const float* __restrict__ g,
                                   const float* __restrict__ bb,
                                   float* __restrict__ scale,
                                   float* __restrict__ shift,
                                   int O, float invCnt) {
  int o = blockIdx.x * blockDim.x + threadIdx.x;
  if (o >= O) return;
  const float mean = stats[o] * invCnt;
  const float var  = stats[128 + o] * invCnt - mean * mean;
  const float sc = g[o] * rsqrtf(var + BN_EPS);
  scale[o] = sc;
  shift[o] = bb[o] - mean * sc;
}

// ---------------------------------------------------------------------------
// Fused normalize + LeakyReLU + max-over-K. One block per point (b,n),
// one thread per channel o. Writes the layer's slice of the (B,256,N)
// output, and (for layers 0/1) the next layer's transposed input plus its
// squared norm (LDS block reduction) -- so no per-layer re-prep is needed.
// ---------------------------------------------------------------------------
__global__ void __launch_bounds__(128)
edge_out_kernel(const float* __restrict__ U, const float* __restrict__ V,
                const int* __restrict__ knn, const float* __restrict__ scale,
                const float* __restrict__ shift, float* __restrict__ out,
                float* __restrict__ xTnext, float* __restrict__ xxnext,
                int O, int N, int B, int coff, int writeNext) {
  __shared__ int   ip[KNN_K];
  __shared__ float sm[128];
  const int bn = blockIdx.x;                      // grid = B*N
  const int o  = threadIdx.x;                     // blockDim = O
  const int b = bn / N, n = bn - b * N;
  if (o < KNN_K) ip[o] = knn[(size_t)bn * KNN_K + o];
  __syncthreads();

  const float* Ub = U + ((size_t)b * O + o) * N;
  const float vv = V[((size_t)b * O + o) * N + n];
  const float sc = scale[o], sh = shift[o];
  float m = -3.0e38f;
#pragma unroll
  for (int k = 0; k < KNN_K; ++k) {
    float y = sc * (Ub[ip[k]] + vv) + sh;
    y = (y > 0.f) ? y : NEG_SLOPE * y;
    m = fmaxf(m, y);
  }
  out[((size_t)b * TOTC + coff + o) * N + n] = m;

  if (writeNext) {
    xTnext[(size_t)bn * O + o] = m;               // next layer's (B,N,CPAD) input
    sm[o] = m * m;
    __syncthreads();
    for (int st = O >> 1; st > 0; st >>= 1) {
      if (o < st) sm[o] += sm[o + st];
      __syncthreads();
    }
    if (o == 0) xxnext[bn] = sm[0];
  }
}

__global__ void zero_kernel(float* __restrict__ p, int n) {
  int i = blockIdx.x * blockDim.x + threadIdx.x;
  if (i < n) p[i] = 0.f;
}

// ---------------------------------------------------------------------------
extern "C" void kernel_launch(void* const* d_in, const int* in_sizes, int n_in,
                              void* d_out, int out_size, void* d_ws, size_t ws_size,
                              hipStream_t stream) {
  (void)in_sizes; (void)n_in; (void)out_size; (void)ws_size;
  const float* x  = (const float*)d_in[0];
  const float* Wt[3] = { (const float*)d_in[1], (const float*)d_in[4], (const float*)d_in[7] };
  const float* gg[3] = { (const float*)d_in[2], (const float*)d_in[5], (const float*)d_in[8] };
  const float* bb[3] = { (const float*)d_in[3], (const float*)d_in[6], (const float*)d_in[9] };
  float* out = (float*)d_out;

  const int B = 4, N = 4096;
  const int Cin_[3]  = { 3, 64, 64 };
  const int Cpad_[3] = { 4, 64, 64 };
  const int Oc_[3]   = { 64, 64, 128 };
  const int coff_[3] = { 0, 64, 128 };

  // Workspace carve-up (~22 MB)
  char* w = (char*)d_ws;
  float* xx   = (float*)w; w += (size_t)B * N * sizeof(float);
  int*   knn  = (int*)w;   w += (size_t)B * N * KNN_K * sizeof(int);
  float* U    = (float*)w; w += (size_t)B * 128 * N * sizeof(float);
  float* V    = (float*)w; w += (size_t)B * 128 * N * sizeof(float);
  float* xT   = (float*)w; w += (size_t)B * N * 64 * sizeof(float);   // (B,N,CPAD)
  float* W1p  = (float*)w; w += (size_t)128 * 64 * sizeof(float);
  float* Wdp  = (float*)w; w += (size_t)128 * 64 * sizeof(float);
  float* stats = (float*)w; w += 256 * sizeof(float);
  float* scale = (float*)w; w += 128 * sizeof(float);
  float* shift = (float*)w; w += 128 * sizeof(float);

  // One-time transpose/pad of the raw input; later layers get xT/xx
  // directly from edge_out_kernel.
  prep_x_kernel<<<(B * N + 255) / 256, 256, 0, stream>>>(x, xT, xx, 3, 4, N, B);

  for (int l = 0; l < 3; ++l) {
    const int Cin = Cin_[l], CPAD = Cpad_[l], O = Oc_[l];

    prep_w_kernel<<<(O * CPAD + 255) / 256, 256, 0, stream>>>(Wt[l], W1p, Wdp, Cin, CPAD, O);

    const int kblocks = (B * (N / 16)) / 8;               // 8 waves / block
    if (CPAD == 4)
      knn_kernel<4><<<kblocks, 256, 0, stream>>>(xT, xx, knn, N, B);
    else
      knn_kernel<64><<<kblocks, 256, 0, stream>>>(xT, xx, knn, N, B);

    const int pblocks = (B * (O / 16) * (N / 16)) / 8;
    if (CPAD == 4)
      ptrans_kernel<4><<<pblocks, 256, 0, stream>>>(xT, W1p, Wdp, U, V, O, N, B);
    else
      ptrans_kernel<64><<<pblocks, 256, 0, stream>>>(xT, W1p, Wdp, U, V, O, N, B);

    zero_kernel<<<1, 256, 0, stream>>>(stats, 256);

    dim3 sgrid(O, 64);                                    // 64*256 == B*N threads/channel
    bn_stats_kernel<<<sgrid, 256, 0, stream>>>(U, V, knn, stats, O, N, B);

    bn_finalize_kernel<<<1, 128, 0, stream>>>(stats, gg[l], bb[l], scale, shift, O,
                                              1.0f / (float)(B * N * KNN_K));

    // Overwrites xT/xx for the next layer (all readers of the current xT/xx
    // have completed earlier in the stream).
    edge_out_kernel<<<B * N, O, 0, stream>>>(U, V, knn, scale, shift, out,
                                             xT, xx, O, N, B, coff_[l], (l < 2) ? 1 : 0);
  }
}